// SacrsnSeedGPT_40948218200575
// MI455X (gfx1250) — compile-verified
//
#include <hip/hip_runtime.h>
#include <hip/hip_bf16.h>
#include <math.h>

// ---------------------------------------------------------------------------
// SacrsnSeedGPT forward, fp32, CDNA5 (gfx1250) WMMA f32 16x16x4 path.
// SIMS loop in the reference is deterministic & identical -> compute one sim.
// GEMM v3: 128x64 block tile, double-buffered LDS B tile, software-pipelined
// chunk-ahead global loads so s_wait_* lands behind 32 WMMAs of compute.
// ---------------------------------------------------------------------------

#define EMBED_D   384
#define HEADS     6
#define HEAD_D    64
#define EXPERTS   4
#define LAYERS    6
#define TBLK      256
#define WINDOW    64
#define VOCAB     32000
#define BATCH     16
#define ROWS      (BATCH * TBLK)      // 4096 token rows
#define D3        (3 * EMBED_D)       // 1152
#define FFN_D     (4 * EMBED_D)       // 1536
#define LN_EPS    1e-5f

#define BM 128                        // block rows   (8 waves x 16)
#define BN 64                         // block cols   (4 wmma tiles per wave)
#define BK 32                         // k-chunk
#define BSTRIDE (BN + 4)              // LDS row pad: half-waves hit disjoint banks

typedef float v2f __attribute__((ext_vector_type(2)));
typedef float v8f __attribute__((ext_vector_type(8)));

// ------------------------------ embedding ----------------------------------
__global__ void embed_kernel(const float* __restrict__ tok,
                             const float* __restrict__ pos,
                             const int*   __restrict__ idx,
                             float* __restrict__ x, long total) {
  long i = (long)blockIdx.x * blockDim.x + threadIdx.x;
  if (i >= total) return;
  int d  = (int)(i % EMBED_D);
  long bt = i / EMBED_D;
  int t  = (int)(bt % TBLK);
  x[i] = tok[(size_t)idx[bt] * EMBED_D + d] + pos[(size_t)t * EMBED_D + d];
}

// ------------------------------ layernorm ----------------------------------
__global__ __launch_bounds__(128)
void ln_kernel(const float* __restrict__ x, const float* __restrict__ w,
               const float* __restrict__ b, float* __restrict__ out) {
  int row  = blockIdx.x;
  int tid  = threadIdx.x;
  int lane = tid & 31, wv = tid >> 5;
  const float* xr = x + (size_t)row * EMBED_D;
  __shared__ float red[4];
  __shared__ float bcast[2];

  float s = 0.f;
  for (int d = tid; d < EMBED_D; d += 128) s += xr[d];
  for (int off = 16; off; off >>= 1) s += __shfl_down(s, off, 32);
  if (lane == 0) red[wv] = s;
  __syncthreads();
  if (tid == 0) bcast[0] = (red[0] + red[1] + red[2] + red[3]) * (1.0f / EMBED_D);
  __syncthreads();
  float mu = bcast[0];

  float v = 0.f;
  for (int d = tid; d < EMBED_D; d += 128) { float t = xr[d] - mu; v += t * t; }
  for (int off = 16; off; off >>= 1) v += __shfl_down(v, off, 32);
  if (lane == 0) red[wv] = v;
  __syncthreads();
  if (tid == 0) bcast[1] = rsqrtf((red[0] + red[1] + red[2] + red[3]) * (1.0f / EMBED_D) + LN_EPS);
  __syncthreads();
  float rs = bcast[1];

  float* orow = out + (size_t)row * EMBED_D;
  for (int d = tid; d < EMBED_D; d += 128)
    orow[d] = (xr[d] - mu) * rs * w[d] + b[d];
}

// ------------------------- WMMA fp32 GEMM core -----------------------------
// C[M,N] = A[M,K] @ W[K,N] (+ bias) with fused epilogues.
// Requirements (all satisfied by this model): M%128==0, N%64==0, K%32==0.
enum { EPI_BIAS = 0, EPI_GELU = 1, EPI_RES = 2, EPI_MOE = 3 };

template <int EPI>
__global__ __launch_bounds__(256)
void gemm_wmma_f32(const float* __restrict__ A, const float* __restrict__ Wm,
                   const float* __restrict__ bias, float* __restrict__ out,
                   const float* __restrict__ extra,  // RES: residual base; MOE: gate+e (stride 4)
                   int M, int N, int K) {
  __shared__ float bs[2][BK * BSTRIDE];   // double-buffered B tile

  const int tid  = threadIdx.x;
  const int wv   = tid >> 5;
  const int lane = tid & 31;
  const int half = lane >> 4;            // 0: lanes 0-15, 1: lanes 16-31
  const int l    = lane & 15;

  const int n0 = blockIdx.x * BN;        // block's N strip
  const int m0 = blockIdx.y * BM + wv * 16;  // this wave's 16 rows

  const int ko = half * 2;               // ISA 32-bit A/B fragment layout
  const float* __restrict__ arow = A + (size_t)(m0 + l) * K + ko;

  // cooperative B loader: 256 threads x 8 elems = BK*BN floats, coalesced rows
  const int bc = tid & 63;               // col within strip
  const int br = tid >> 6;               // row (0..3), step 4 per iter
  const float* __restrict__ bbase = Wm + (size_t)br * N + n0 + bc;

  v8f acc[4];
#pragma unroll
  for (int t = 0; t < 4; ++t) acc[t] = (v8f){0.f,0.f,0.f,0.f,0.f,0.f,0.f,0.f};

  // ---- prologue: stage chunk 0 into bs[0], preload A fragments ----
#pragma unroll
  for (int i = 0; i < 8; ++i)
    bs[0][(br + 4 * i) * BSTRIDE + bc] = bbase[(size_t)(4 * i) * N];
  v2f areg[8];
#pragma unroll
  for (int j = 0; j < 8; ++j)
    areg[j] = *(const v2f*)(arow + 4 * j);
  __syncthreads();

  int buf = 0;
  for (int kc = 0; kc < K; kc += BK) {
    const bool more = (kc + BK) < K;

    // ---- issue next chunk's global loads (latency hidden by compute) ----
    float breg[8];
    v2f  anext[8];
    if (more) {
      const float* __restrict__ bn = bbase + (size_t)(kc + BK) * N;
#pragma unroll
      for (int i = 0; i < 8; ++i)
        breg[i] = bn[(size_t)(4 * i) * N];
#pragma unroll
      for (int j = 0; j < 8; ++j)
        anext[j] = *(const v2f*)(arow + kc + BK + 4 * j);
    }

    // ---- 32 WMMAs: 8 k-steps x 4 n-subtiles, A reused across subtiles ----
    const float* __restrict__ bsb = bs[buf];
#pragma unroll
    for (int j = 0; j < 8; ++j) {
      const int kk = 4 * j + ko;
      const float* __restrict__ b0 = &bsb[kk * BSTRIDE + l];
      const float* __restrict__ b1 = &bsb[(kk + 1) * BSTRIDE + l];
#pragma unroll
      for (int t = 0; t < 4; ++t) {
        v2f bv;
        bv.x = b0[16 * t];
        bv.y = b1[16 * t];
        acc[t] = __builtin_amdgcn_wmma_f32_16x16x4_f32(false, areg[j], false, bv,
                                                       (short)0, acc[t], false, false);
      }
    }

    // ---- spill staged registers into the other LDS buffer, flip ----
    if (more) {
#pragma unroll
      for (int i = 0; i < 8; ++i)
        bs[buf ^ 1][(br + 4 * i) * BSTRIDE + bc] = breg[i];
#pragma unroll
      for (int j = 0; j < 8; ++j)
        areg[j] = anext[j];
      __syncthreads();
      buf ^= 1;
    }
  }

  // ---- fused epilogue over the wave's 4 sub-tiles ----
#pragma unroll
  for (int t = 0; t < 4; ++t) {
    const int col = n0 + 16 * t + l;
    const float bval = bias ? bias[col] : 0.f;
#pragma unroll
    for (int r = 0; r < 8; ++r) {
      const int row = m0 + r + half * 8;   // C layout: lanes 16-31 hold M=r+8
      const size_t o = (size_t)row * N + col;
      float v = acc[t][r] + bval;
      if constexpr (EPI == EPI_GELU) {
        out[o] = 0.5f * v * (1.0f + erff(v * 0.70710678118654752440f));
      } else if constexpr (EPI == EPI_RES) {
        out[o] = extra[o] + v;             // residual add (extra may alias out)
      } else if constexpr (EPI == EPI_MOE) {
        out[o] += extra[(size_t)row * EXPERTS] * v;  // gated accumulate
      } else {
        out[o] = v;
      }
    }
  }
}

// --------------------- windowed causal attention ---------------------------
// One block (96 threads) per (b, h, q). <= 65 valid keys.
__global__ __launch_bounds__(96)
void attn_kernel(const float* __restrict__ qkv, float* __restrict__ y) {
  const int flat = blockIdx.x;
  const int qp = flat % TBLK;
  const int h  = (flat / TBLK) % HEADS;
  const int b  = flat / (TBLK * HEADS);
  const int tid = threadIdx.x;

  __shared__ float qs[HEAD_D];
  __shared__ float sc[WINDOW + 1];
  __shared__ float invsum;

  const float* qrow = qkv + (size_t)(b * TBLK + qp) * D3 + h * HEAD_D;
  if (tid < HEAD_D) qs[tid] = qrow[tid];
  __syncthreads();

  int kstart = qp - WINDOW; if (kstart < 0) kstart = 0;
  const int nk = qp - kstart + 1;

  if (tid < nk) {
    const float* krow = qkv + (size_t)(b * TBLK + kstart + tid) * D3 + EMBED_D + h * HEAD_D;
    float s = 0.f;
#pragma unroll 8
    for (int d = 0; d < HEAD_D; ++d) s += qs[d] * krow[d];
    sc[tid] = s * 0.125f;                 // 1/sqrt(64)
  }
  __syncthreads();

  if (tid == 0) {
    float m = -1e30f;
    for (int j = 0; j < nk; ++j) m = fmaxf(m, sc[j]);
    float ssum = 0.f;
    for (int j = 0; j < nk; ++j) { float e = expf(sc[j] - m); sc[j] = e; ssum += e; }
    invsum = 1.0f / ssum;
  }
  __syncthreads();

  if (tid < HEAD_D) {
    float accv = 0.f;
    for (int j = 0; j < nk; ++j) {
      const float* vrow = qkv + (size_t)(b * TBLK + kstart + j) * D3 + 2 * EMBED_D + h * HEAD_D;
      accv += sc[j] * vrow[tid];
    }
    y[(size_t)(b * TBLK + qp) * EMBED_D + h * HEAD_D + tid] = accv * invsum;
  }
}

// --------------------------- MoE gate (E=4) --------------------------------
__global__ __launch_bounds__(32)
void gate_kernel(const float* __restrict__ h2, const float* __restrict__ gw,
                 const float* __restrict__ gb, float* __restrict__ gate) {
  const int row = blockIdx.x;
  const int lane = threadIdx.x;
  const float* xr = h2 + (size_t)row * EMBED_D;
  float acc[EXPERTS] = {0.f, 0.f, 0.f, 0.f};
  for (int d = lane; d < EMBED_D; d += 32) {
    const float xv = xr[d];
#pragma unroll
    for (int e = 0; e < EXPERTS; ++e) acc[e] += xv * gw[d * EXPERTS + e];
  }
#pragma unroll
  for (int e = 0; e < EXPERTS; ++e)
    for (int off = 16; off; off >>= 1) acc[e] += __shfl_down(acc[e], off, 32);
  if (lane == 0) {
    float v[EXPERTS], m = -1e30f;
    for (int e = 0; e < EXPERTS; ++e) { v[e] = acc[e] + gb[e]; m = fmaxf(m, v[e]); }
    float s = 0.f;
    for (int e = 0; e < EXPERTS; ++e) { v[e] = expf(v[e] - m); s += v[e]; }
    for (int e = 0; e < EXPERTS; ++e) gate[(size_t)row * EXPERTS + e] = v[e] / s;
  }
}

// ------------------------------- launch ------------------------------------
extern "C" void kernel_launch(void* const* d_in, const int* in_sizes, int n_in,
                              void* d_out, int out_size, void* d_ws, size_t ws_size,
                              hipStream_t stream) {
  (void)in_sizes; (void)n_in; (void)out_size; (void)ws_size;

  // setup_inputs() dict order (depth-first):
  const float* tok_emb = (const float*)d_in[0];
  const float* pos_emb = (const float*)d_in[1];
  const float* ln1_w   = (const float*)d_in[2];
  const float* ln1_b   = (const float*)d_in[3];
  const float* attn_w  = (const float*)d_in[4];
  const float* attn_b  = (const float*)d_in[5];
  const float* proj_w  = (const float*)d_in[6];
  const float* proj_b  = (const float*)d_in[7];
  const float* ln2_w   = (const float*)d_in[8];
  const float* ln2_b   = (const float*)d_in[9];
  const float* gate_w  = (const float*)d_in[10];
  const float* gate_b  = (const float*)d_in[11];
  const float* e1_w    = (const float*)d_in[12];
  const float* e1_b    = (const float*)d_in[13];
  const float* e2_w    = (const float*)d_in[14];
  const float* e2_b    = (const float*)d_in[15];
  const float* lnf_w   = (const float*)d_in[16];
  const float* lnf_b   = (const float*)d_in[17];
  const float* head_w  = (const float*)d_in[18];
  const float* head_b  = (const float*)d_in[19];
  const int*   idx     = (const int*)  d_in[20];

  float* logits = (float*)d_out;

  // Workspace layout (floats): ~69 MB total.
  float* ws   = (float*)d_ws;
  float* x    = ws;                                   // [4096, 384]
  float* h    = x    + (size_t)ROWS * EMBED_D;        // [4096, 384]
  float* qkv  = h    + (size_t)ROWS * EMBED_D;        // [4096, 1152]
  float* y    = qkv  + (size_t)ROWS * D3;             // [4096, 384]
  float* h2   = y    + (size_t)ROWS * EMBED_D;        // [4096, 384]
  float* gate = h2   + (size_t)ROWS * EMBED_D;        // [4096, 4]
  float* hid  = gate + (size_t)ROWS * EXPERTS;        // [4096, 1536]

  // ---- embedding ----
  {
    long total = (long)ROWS * EMBED_D;
    embed_kernel<<<(int)((total + 255) / 256), 256, 0, stream>>>(
        tok_emb, pos_emb, idx, x, total);
  }

  // ---- transformer layers (single sim; SIMS are identical) ----
  for (int L = 0; L < LAYERS; ++L) {
    // LN1
    ln_kernel<<<ROWS, 128, 0, stream>>>(x, ln1_w + (size_t)L * EMBED_D,
                                        ln1_b + (size_t)L * EMBED_D, h);
    // QKV GEMM: [4096,384] @ [384,1152] + b
    gemm_wmma_f32<EPI_BIAS><<<dim3(D3 / BN, ROWS / BM), 256, 0, stream>>>(
        h, attn_w + (size_t)L * EMBED_D * D3, attn_b + (size_t)L * D3,
        qkv, nullptr, ROWS, D3, EMBED_D);
    // windowed causal attention
    attn_kernel<<<BATCH * HEADS * TBLK, 96, 0, stream>>>(qkv, y);
    // proj GEMM + residual: x = x + y @ proj_w + b
    gemm_wmma_f32<EPI_RES><<<dim3(EMBED_D / BN, ROWS / BM), 256, 0, stream>>>(
        y, proj_w + (size_t)L * EMBED_D * EMBED_D, proj_b + (size_t)L * EMBED_D,
        x, x, ROWS, EMBED_D, EMBED_D);
    // LN2
    ln_kernel<<<ROWS, 128, 0, stream>>>(x, ln2_w + (size_t)L * EMBED_D,
                                        ln2_b + (size_t)L * EMBED_D, h2);
    // gate = softmax(h2 @ gate_w + gate_b)
    gate_kernel<<<ROWS, 32, 0, stream>>>(h2, gate_w + (size_t)L * EMBED_D * EXPERTS,
                                         gate_b + (size_t)L * EXPERTS, gate);
    // MoE experts: x += gate[:,e] * (gelu(h2 @ e1 + b1) @ e2 + b2)
    for (int e = 0; e < EXPERTS; ++e) {
      const size_t le = (size_t)L * EXPERTS + e;
      gemm_wmma_f32<EPI_GELU><<<dim3(FFN_D / BN, ROWS / BM), 256, 0, stream>>>(
          h2, e1_w + le * EMBED_D * FFN_D, e1_b + le * FFN_D,
          hid, nullptr, ROWS, FFN_D, EMBED_D);
      gemm_wmma_f32<EPI_MOE><<<dim3(EMBED_D / BN, ROWS / BM), 256, 0, stream>>>(
          hid, e2_w + le * FFN_D * EMBED_D, e2_b + le * EMBED_D,
          x, gate + e, ROWS, EMBED_D, FFN_D);
    }
  }

  // ---- final LN + vocab head ----
  ln_kernel<<<ROWS, 128, 0, stream>>>(x, lnf_w, lnf_b, h);
  gemm_wmma_f32<EPI_BIAS><<<dim3(VOCAB / BN, ROWS / BM), 256, 0, stream>>>(
      h, head_w, head_b, logits, nullptr, ROWS, VOCAB, EMBED_D);
}